// FastNDCG_78451872629496
// MI455X (gfx1250) — compile-verified
//
#include <hip/hip_runtime.h>
#include <hip/hip_bf16.h>
#include <stdint.h>

#define NUM_USERS 65536
#define KCUT 10

typedef __attribute__((ext_vector_type(2))) float v2f;
typedef __attribute__((ext_vector_type(8))) float v8f;

// ---------- helpers ----------
__device__ __forceinline__ unsigned sortable_f32(float f) {
    unsigned u = __float_as_uint(f);
    return (u & 0x80000000u) ? ~u : (u | 0x80000000u);
}

// ---------- pass 0: zero per-user counters ----------
__global__ void k_zero(int* __restrict__ counts) {
    int i = blockIdx.x * blockDim.x + threadIdx.x;
    if (i < NUM_USERS) counts[i] = 0;
}

// ---------- pass 1: per-user histogram ----------
__global__ void k_count(const int* __restrict__ idx, int* __restrict__ counts, int n) {
    int i = blockIdx.x * blockDim.x + threadIdx.x;
    if (i < n) atomicAdd(&counts[idx[i]], 1);
}

// ---------- pass 2: exclusive scan of 65536 counts (one 1024-thread block) ----------
__global__ void __launch_bounds__(1024) k_scan(const int* __restrict__ counts,
                                               int* __restrict__ starts,
                                               int* __restrict__ cursors) {
    __shared__ int part[1024];
    const int tid = threadIdx.x;
    const int base = tid * 64;
    int s = 0;
    for (int j = 0; j < 64; ++j) s += counts[base + j];
    part[tid] = s;
    __syncthreads();
    // Hillis-Steele inclusive scan (read, sync, write, sync)
    for (int off = 1; off < 1024; off <<= 1) {
        int v = (tid >= off) ? part[tid - off] : 0;
        __syncthreads();
        part[tid] += v;
        __syncthreads();
    }
    int run = part[tid] - s;  // exclusive prefix at chunk start
    for (int j = 0; j < 64; ++j) {
        starts[base + j]  = run;
        cursors[base + j] = run;
        run += counts[base + j];
    }
}

// ---------- pass 3: scatter packed keys into per-user buckets ----------
__global__ void k_scatter(const float* __restrict__ pred, const float* __restrict__ tgt,
                          const int* __restrict__ idx, int* __restrict__ cursors,
                          unsigned* __restrict__ buf, int n) {
    int i = blockIdx.x * blockDim.x + threadIdx.x;
    if (i < n) {
        int u = idx[i];
        int pos = atomicAdd(&cursors[u], 1);
        unsigned key = (sortable_f32(pred[i]) & ~1u) | (tgt[i] > 0.5f ? 1u : 0u);
        buf[pos] = key;
    }
}

// ---------- pass 4: one wave32 per user: top-10 extraction + NDCG ----------
__global__ void k_user(const unsigned* __restrict__ buf, const int* __restrict__ starts,
                       const int* __restrict__ counts,
                       float* __restrict__ ndcg, float* __restrict__ valid) {
    const int lane = threadIdx.x & 31;
    const int wave = threadIdx.x >> 5;
    const int u = blockIdx.x * (blockDim.x >> 5) + wave;
    if (u >= NUM_USERS) return;  // wave-uniform

    const int st  = starts[u];
    const int cnt = counts[u];

    // per-lane register-resident sorted (desc) top-10; 0 is a safe sentinel
    unsigned loc[KCUT];
#pragma unroll
    for (int j = 0; j < KCUT; ++j) loc[j] = 0u;

    int tl = 0;  // local target count
    for (int j = lane; j < cnt; j += 32) {
        unsigned key = buf[st + j];
        tl += (int)(key & 1u);
        if (key > loc[KCUT - 1]) {
            int p = KCUT - 1;
#pragma unroll
            for (int q = KCUT - 2; q >= 0; --q) {
                if (key > loc[q]) { loc[q + 1] = loc[q]; p = q; }
            }
            loc[p] = key;
        }
    }

    // wave-wide total target count (integer, order-independent)
    for (int off = 16; off > 0; off >>= 1) tl += __shfl_xor(tl, off, 32);

    const float disc[KCUT] = {
        1.0f, 0.6309297535714575f, 0.5f, 0.43067655807339306f, 0.38685280723454163f,
        0.35620718710802223f, 0.3333333333333333f, 0.31546487678572877f,
        0.30102999566398114f, 0.2890648263178878f
    };

    // 10 rounds of wave-wide max extraction (static-indexed list shift)
    float dcg = 0.f;
#pragma unroll
    for (int r = 0; r < KCUT; ++r) {
        unsigned my = loc[0];
        unsigned mx = my;
        for (int off = 16; off > 0; off >>= 1) {
            unsigned o = __shfl_xor(mx, off, 32);
            mx = (o > mx) ? o : mx;
        }
        int cand = (my == mx) ? lane : 32;
        for (int off = 16; off > 0; off >>= 1) {
            int o = __shfl_xor(cand, off, 32);
            cand = (o < cand) ? o : cand;
        }
        if (lane == cand) {  // winner pops head of its list
#pragma unroll
            for (int q = 0; q < KCUT - 1; ++q) loc[q] = loc[q + 1];
            loc[KCUT - 1] = 0u;
        }
        dcg += (mx & 1u) ? disc[r] : 0.f;  // sentinel (mx==0) contributes 0
    }

    if (lane == 0) {
        int m = tl < KCUT ? tl : KCUT;
        float idcg = 0.f;
#pragma unroll
        for (int r = 0; r < KCUT; ++r) idcg += (r < m) ? disc[r] : 0.f;
        bool ok = (tl > 0);
        ndcg[u]  = ok ? (dcg / idcg) : 0.f;
        valid[u] = ok ? 1.f : 0.f;
    }
}

// ---------- pass 5: fixed-order tree reduction 65536 -> 32 partials ----------
__global__ void k_reduce1(const float* __restrict__ ndcg, const float* __restrict__ valid,
                          float* __restrict__ pn, float* __restrict__ pv) {
    __shared__ float sn[256], sv[256];
    const int t = threadIdx.x;
    const int base = blockIdx.x * 2048;
    float a = 0.f, b = 0.f;
    for (int j = 0; j < 8; ++j) {
        a += ndcg[base + t + 256 * j];
        b += valid[base + t + 256 * j];
    }
    sn[t] = a; sv[t] = b;
    __syncthreads();
    for (int off = 128; off > 0; off >>= 1) {
        if (t < off) { sn[t] += sn[t + off]; sv[t] += sv[t + off]; }
        __syncthreads();
    }
    if (t == 0) { pn[blockIdx.x] = sn[0]; pv[blockIdx.x] = sv[0]; }
}

// ---------- pass 6: WMMA ones-matmul reduction of the 32 partials ----------
// D = A(16x4) * ones(4x16): sum of ALL D elements = 16 * sum(A elements),
// independent of exact lane layout. One wave32, EXEC all ones, no divergence
// before the WMMAs.
__global__ void k_final(const float* __restrict__ pn, const float* __restrict__ pv,
                        float* __restrict__ out) {
    const int lane = threadIdx.x;  // 32 threads
    v2f an, av, ones;
    an[0] = pn[lane]; an[1] = 0.f;
    av[0] = pv[lane]; av[1] = 0.f;
    ones[0] = 1.f; ones[1] = 1.f;
    v8f cz = {};
    v8f dn = __builtin_amdgcn_wmma_f32_16x16x4_f32(false, an, false, ones, (short)0, cz, false, false);
    v8f dv = __builtin_amdgcn_wmma_f32_16x16x4_f32(false, av, false, ones, (short)0, cz, false, false);
    float sn = 0.f, sv = 0.f;
#pragma unroll
    for (int g = 0; g < 8; ++g) { sn += dn[g]; sv += dv[g]; }
    for (int off = 16; off > 0; off >>= 1) {
        sn += __shfl_xor(sn, off, 32);
        sv += __shfl_xor(sv, off, 32);
    }
    sn *= (1.f / 16.f);
    sv *= (1.f / 16.f);   // exact: integer-valued counts, power-of-two scale
    if (lane == 0) {
        out[0] = (sv > 0.f) ? (sn / fmaxf(sv, 1.f)) : 0.f;
    }
}

// ---------- launcher ----------
extern "C" void kernel_launch(void* const* d_in, const int* in_sizes, int n_in,
                              void* d_out, int out_size, void* d_ws, size_t ws_size,
                              hipStream_t stream) {
    (void)n_in; (void)out_size; (void)ws_size;
    const float* pred = (const float*)d_in[0];
    const float* tgt  = (const float*)d_in[1];
    const int*   idx  = (const int*)d_in[2];
    const int n = in_sizes[0];

    char* ws = (char*)d_ws;
    int*   counts  = (int*)(ws);
    int*   starts  = (int*)(ws + (size_t)NUM_USERS * 4);
    int*   cursors = (int*)(ws + (size_t)NUM_USERS * 8);
    float* ndcg    = (float*)(ws + (size_t)NUM_USERS * 12);
    float* valid   = (float*)(ws + (size_t)NUM_USERS * 16);
    float* pn      = (float*)(ws + (size_t)NUM_USERS * 20);
    float* pv      = pn + 32;
    unsigned* buf  = (unsigned*)(ws + (size_t)NUM_USERS * 20 + 4096);  // n * 4 bytes

    k_zero<<<(NUM_USERS + 255) / 256, 256, 0, stream>>>(counts);
    k_count<<<(n + 255) / 256, 256, 0, stream>>>(idx, counts, n);
    k_scan<<<1, 1024, 0, stream>>>(counts, starts, cursors);
    k_scatter<<<(n + 255) / 256, 256, 0, stream>>>(pred, tgt, idx, cursors, buf, n);
    k_user<<<NUM_USERS / 8, 256, 0, stream>>>(buf, starts, counts, ndcg, valid);
    k_reduce1<<<32, 256, 0, stream>>>(ndcg, valid, pn, pv);
    k_final<<<1, 32, 0, stream>>>(pn, pv, (float*)d_out);
}